// CtpnTarget_21131239096447
// MI455X (gfx1250) — compile-verified
//
#include <hip/hip_runtime.h>
#include <hip/hip_bf16.h>
#include <stdint.h>
#include <math.h>

// Problem constants (from the reference)
#define B_    16
#define G_    50
#define A_    81920
#define T_    128
#define P_    64
#define NNEG_ 64

// Implementation constants
#define PCAP   4096      // capped positive-candidate list per image
#define NCAP   4096      // capped negative-candidate list per image
#define NEGTH  0.99f     // hash-score threshold for negative candidates (~800 expected of ~80k)
#define CHUNK  128       // anchors per LDS chunk per block (8 waves x 16 anchors)
#define NCHUNK 4
#define BLKA   (CHUNK*NCHUNK)   // 512 anchors per block

typedef float v2f __attribute__((ext_vector_type(2)));
typedef float v8f __attribute__((ext_vector_type(8)));

// ---------- CDNA5 async global->LDS helpers (inline asm; ASYNCcnt tracked) ----------
__device__ __forceinline__ unsigned lds_off_of(const void* p) {
  // flat LDS pointers carry the byte offset in the low 32 bits
  return (unsigned)(uintptr_t)p;
}
__device__ __forceinline__ void async_load_b64(unsigned ldsOff, const void* gptr) {
  asm volatile("global_load_async_to_lds_b64 %0, %1, off"
               :: "v"(ldsOff), "v"((unsigned long long)(uintptr_t)gptr)
               : "memory");
}
__device__ __forceinline__ void async_wait0() { asm volatile("s_wait_asynccnt 0x0" ::: "memory"); }
__device__ __forceinline__ void async_wait1() { asm volatile("s_wait_asynccnt 0x1" ::: "memory"); }

// fast reciprocal (v_rcp_f32) — used identically in both IoU passes so
// bitwise equality between pass1 max and pass2 recomputation is preserved
__device__ __forceinline__ float fast_rcp(float x) { return __builtin_amdgcn_rcpf(x); }

// ---------- deterministic hash "PRNG" ----------
__device__ __forceinline__ unsigned hash32(unsigned x) {
  x ^= x >> 16; x *= 0x7feb352du;
  x ^= x >> 15; x *= 0x846ca68bu;
  x ^= x >> 16; return x;
}
__device__ __forceinline__ float hash01(unsigned x) {
  return (float)(hash32(x) & 0xFFFFFFu) * (1.0f / 16777216.0f);
}

// ---------- output offsets (floats) ----------
#define OFF_D  0
#define OFF_C  (B_*T_*3)
#define OFF_I  (OFF_C + B_*T_*2)
#define OFF_SD (OFF_I + B_*T_*2)
#define OFF_SI (OFF_SD + B_*G_*3)
#define OFF_GN (OFF_SI + B_*G_*3)
#define OFF_PN (OFF_GN + B_)
#define OFF_NN (OFF_PN + B_)

// =====================================================================
__global__ __launch_bounds__(256) void k_init(float* out, int outN,
                                              unsigned* gmax,
                                              unsigned long long* lE,
                                              unsigned long long* rE,
                                              int* pC, int* nC, int* nT) {
  int i = blockIdx.x * blockDim.x + threadIdx.x;
  int stride = gridDim.x * blockDim.x;
  for (int k = i; k < outN; k += stride) out[k] = 0.f;
  for (int k = i; k < B_ * 64; k += stride) { gmax[k] = 0u; lE[k] = ~0ull; rE[k] = 0ull; }
  for (int k = i; k < B_; k += stride) { pC[k] = 0; nC[k] = 0; nT[k] = 0; }
}

// =====================================================================
// Pass 1: per-GT max IoU over all anchors (float-bits u32 atomics).
// IoU denominators (area_g + area_a) are produced by V_WMMA_F32_16X16X4_F32.
__global__ __launch_bounds__(256) void k_pass1(const float* __restrict__ gtB,
                                               const float* __restrict__ anchors,
                                               unsigned* __restrict__ gGmax) {
  const int b = blockIdx.x;
  const int tt = threadIdx.x;
  const int w = tt >> 5, lane = tt & 31, al = lane & 15, hf = lane >> 4;

  __shared__ float    s_gb[64][4];
  __shared__ float    s_ag[64];
  __shared__ unsigned s_gmax[64];
  __shared__ float4   s_an[2][CHUNK];

  if (tt < 64) {
    float y1 = 0, x1 = 0, y2 = 0, x2 = 0;
    if (tt < G_) {
      const float* r = gtB + ((size_t)b * G_ + tt) * 5;
      y1 = r[0]; x1 = r[1]; y2 = r[2]; x2 = r[3];
    }
    s_gb[tt][0] = y1; s_gb[tt][1] = x1; s_gb[tt][2] = y2; s_gb[tt][3] = x2;
    s_ag[tt] = (x2 - x1) * (y2 - y1);
    s_gmax[tt] = 0u;
  }
  __syncthreads();

  // A fragments: 16x4 f32 per GT tile: col0 = area_g, col1 = 1
  v2f aA[4];
#pragma unroll
  for (int t4 = 0; t4 < 4; ++t4) {
    aA[t4][0] = (hf == 0) ? s_ag[t4 * 16 + al] : 0.f;
    aA[t4][1] = (hf == 0) ? 1.f : 0.f;
  }

  const int blockBase = blockIdx.y * BLKA;
  const float* anchB = anchors + (size_t)b * A_ * 4;

  // issue chunk 0 (wave loads its own 16 anchors: 32 lanes x 8B)
  {
    unsigned lds = lds_off_of(&s_an[0][w * 16]) + lane * 8;
    const char* g = (const char*)(anchB + (size_t)(blockBase + w * 16) * 4) + lane * 8;
    async_load_b64(lds, g);
  }

  for (int it = 0; it < NCHUNK; ++it) {
    if (it + 1 < NCHUNK) {
      unsigned lds = lds_off_of(&s_an[(it + 1) & 1][w * 16]) + lane * 8;
      const char* g = (const char*)(anchB + (size_t)(blockBase + (it + 1) * CHUNK + w * 16) * 4) + lane * 8;
      async_load_b64(lds, g);
      async_wait1();
    } else {
      async_wait0();
    }

    float4 pa = s_an[it & 1][w * 16 + al];
    float areaA = (pa.w - pa.y) * (pa.z - pa.x);
    v2f bB; bB[0] = (hf == 0) ? 1.f : 0.f; bB[1] = (hf == 0) ? areaA : 0.f;

#pragma unroll
    for (int t4 = 0; t4 < 4; ++t4) {
      v8f c = {0.f, 0.f, 0.f, 0.f, 0.f, 0.f, 0.f, 0.f};
      // D[g][a] = area_g[g] + area_a[a]
      v8f dd = __builtin_amdgcn_wmma_f32_16x16x4_f32(false, aA[t4], false, bB,
                                                     (short)0, c, false, false);
#pragma unroll
      for (int r = 0; r < 8; ++r) {
        int g = t4 * 16 + hf * 8 + r;
        float iw = fminf(s_gb[g][3], pa.w) - fmaxf(s_gb[g][1], pa.y); iw = fmaxf(iw, 0.f);
        float ih = fminf(s_gb[g][2], pa.z) - fmaxf(s_gb[g][0], pa.x); ih = fmaxf(ih, 0.f);
        float inter = iw * ih;
        // untagged GT boxes are all-zero -> inter==0 -> iou==0 (den = areaA > 0)
        float iou = inter * fast_rcp(fmaxf(dd[r] - inter, 1e-12f));
        if (iou > 0.f) atomicMax(&s_gmax[g], __float_as_uint(iou));
      }
    }
  }
  __syncthreads();
  if (tt < G_) atomicMax(&gGmax[b * 64 + tt], s_gmax[tt]);
}

// =====================================================================
// Pass 2: classify anchors, build candidate lists, side-anchor atomics.
__global__ __launch_bounds__(256) void k_pass2(const float* __restrict__ gtB,
                                               const float* __restrict__ anchors,
                                               const unsigned* __restrict__ gGmax,
                                               unsigned long long* __restrict__ lE,
                                               unsigned long long* __restrict__ rE,
                                               int* __restrict__ posCnt, float* __restrict__ posS, unsigned* __restrict__ posI,
                                               int* __restrict__ negCnt, int* __restrict__ negTot,
                                               float* __restrict__ negS, unsigned* __restrict__ negI) {
  const int b = blockIdx.x;
  const int tt = threadIdx.x;
  const int w = tt >> 5, lane = tt & 31, al = lane & 15, hf = lane >> 4;

  __shared__ float    s_gb[64][4];
  __shared__ float    s_tag[64];
  __shared__ float    s_ag[64];
  __shared__ unsigned s_gmax[64];
  __shared__ float4   s_an[2][CHUNK];

  if (tt < 64) {
    float y1 = 0, x1 = 0, y2 = 0, x2 = 0, tg = 0;
    if (tt < G_) {
      const float* r = gtB + ((size_t)b * G_ + tt) * 5;
      y1 = r[0]; x1 = r[1]; y2 = r[2]; x2 = r[3]; tg = (r[4] > 0.f) ? 1.f : 0.f;
    }
    s_gb[tt][0] = y1; s_gb[tt][1] = x1; s_gb[tt][2] = y2; s_gb[tt][3] = x2;
    s_tag[tt] = tg; s_ag[tt] = (x2 - x1) * (y2 - y1);
    s_gmax[tt] = gGmax[b * 64 + tt];
  }
  __syncthreads();

  v2f aA[4];
#pragma unroll
  for (int t4 = 0; t4 < 4; ++t4) {
    aA[t4][0] = (hf == 0) ? s_ag[t4 * 16 + al] : 0.f;
    aA[t4][1] = (hf == 0) ? 1.f : 0.f;
  }

  const int blockBase = blockIdx.y * BLKA;
  const float* anchB = anchors + (size_t)b * A_ * 4;

  {
    unsigned lds = lds_off_of(&s_an[0][w * 16]) + lane * 8;
    const char* g = (const char*)(anchB + (size_t)(blockBase + w * 16) * 4) + lane * 8;
    async_load_b64(lds, g);
  }

  for (int it = 0; it < NCHUNK; ++it) {
    if (it + 1 < NCHUNK) {
      unsigned lds = lds_off_of(&s_an[(it + 1) & 1][w * 16]) + lane * 8;
      const char* g = (const char*)(anchB + (size_t)(blockBase + (it + 1) * CHUNK + w * 16) * 4) + lane * 8;
      async_load_b64(lds, g);
      async_wait1();
    } else {
      async_wait0();
    }

    float4 pa = s_an[it & 1][w * 16 + al];
    const int aIdx = blockBase + it * CHUNK + w * 16 + al;
    float areaA = (pa.w - pa.y) * (pa.z - pa.x);
    v2f bB; bB[0] = (hf == 0) ? 1.f : 0.f; bB[1] = (hf == 0) ? areaA : 0.f;

    float iouv[32];
    float amaxh = 0.f;

#pragma unroll
    for (int t4 = 0; t4 < 4; ++t4) {
      v8f c = {0.f, 0.f, 0.f, 0.f, 0.f, 0.f, 0.f, 0.f};
      v8f dd = __builtin_amdgcn_wmma_f32_16x16x4_f32(false, aA[t4], false, bB,
                                                     (short)0, c, false, false);
#pragma unroll
      for (int r = 0; r < 8; ++r) {
        int g = t4 * 16 + hf * 8 + r;
        float iw = fminf(s_gb[g][3], pa.w) - fmaxf(s_gb[g][1], pa.y); iw = fmaxf(iw, 0.f);
        float ih = fminf(s_gb[g][2], pa.z) - fmaxf(s_gb[g][0], pa.x); ih = fmaxf(ih, 0.f);
        float inter = iw * ih;
        float iou = inter * fast_rcp(fmaxf(dd[r] - inter, 1e-12f));
        iouv[t4 * 8 + r] = iou;
        amaxh = fmaxf(amaxh, iou);
      }
    }

    // combine the two half-lanes covering the same anchor
    float amax = fmaxf(amaxh, __shfl_xor(amaxh, 16, 32));
    float athr = (amax >= 0.7f) ? amax : 1.0f;

    int anyp = 0;
#pragma unroll
    for (int t4 = 0; t4 < 4; ++t4) {
#pragma unroll
      for (int r = 0; r < 8; ++r) {
        int g = t4 * 16 + hf * 8 + r;
        float iou = iouv[t4 * 8 + r];
        bool pos = (s_tag[g] > 0.f) &&
                   ((__float_as_uint(iou) == s_gmax[g]) || (iou == athr));
        if (pos) {
          anyp = 1;
          unsigned pk = (unsigned)g * (unsigned)A_ + (unsigned)aIdx;
          float sc = hash01(((unsigned)(b * G_ + g) * 0x9E3779B9u) ^
                            ((unsigned)aIdx * 0x85ebca6bu));
          int id = atomicAdd(&posCnt[b], 1);
          if (id < PCAP) { posS[b * PCAP + id] = sc; posI[b * PCAP + id] = pk; }
          unsigned xb = __float_as_uint(pa.y);  // anchor x1 >= 0 -> order-preserving bits
          atomicMin(&lE[b * 64 + g],
                    ((unsigned long long)xb << 32) | (unsigned long long)(unsigned)aIdx);
          atomicMax(&rE[b * 64 + g],
                    ((unsigned long long)xb << 32) |
                    (unsigned long long)(0xFFFFFFFFu - (unsigned)aIdx));
        }
      }
    }
    int anyAll = anyp | __shfl_xor(anyp, 16, 32);
    if (hf == 0) {
      if (amax < 0.5f && !anyAll) {
        atomicAdd(&negTot[b], 1);
        float sc = hash01(((unsigned)b * 0x01000193u) ^ ((unsigned)aIdx * 0xC2B2AE35u));
        if (sc > NEGTH) {
          int id = atomicAdd(&negCnt[b], 1);
          if (id < NCAP) { negS[b * NCAP + id] = sc; negI[b * NCAP + id] = (unsigned)aIdx; }
        }
      }
    }
  }
}

// =====================================================================
__device__ int topk_select(float* sS, unsigned* sI, int n, int K,
                           float* rS, unsigned* rI, unsigned* sel, int* sCnt) {
  const int t = threadIdx.x;
  if (t == 0) *sCnt = 0;
  __syncthreads();
  for (int k = 0; k < K; ++k) {
    float bs = -1.f; unsigned bi = 0xFFFFFFFFu;
    for (int i = t; i < n; i += 256) { float v = sS[i]; if (v > bs) { bs = v; bi = (unsigned)i; } }
    rS[t] = bs; rI[t] = bi;
    __syncthreads();
    for (int off = 128; off > 0; off >>= 1) {
      if (t < off) { if (rS[t + off] > rS[t]) { rS[t] = rS[t + off]; rI[t] = rI[t + off]; } }
      __syncthreads();
    }
    if (t == 0) {
      if (rS[0] >= 0.f) { sel[k] = sI[rI[0]]; sS[rI[0]] = -2.f; (*sCnt)++; }
      else sel[k] = 0xFFFFFFFFu;
    }
    __syncthreads();
  }
  return *sCnt;
}

__global__ __launch_bounds__(256) void k_final(const float* __restrict__ gtB,
                                               const float* __restrict__ gtC,
                                               const float* __restrict__ anch,
                                               const int* __restrict__ vIdx,
                                               const unsigned long long* __restrict__ lE,
                                               const unsigned long long* __restrict__ rE,
                                               const int* __restrict__ pC, const float* __restrict__ pS, const unsigned* __restrict__ pI,
                                               const int* __restrict__ nC, const int* __restrict__ nT,
                                               const float* __restrict__ nS, const unsigned* __restrict__ nI,
                                               float* __restrict__ out) {
  const int b = blockIdx.x;
  const int t = threadIdx.x;

  __shared__ float    sS[PCAP];
  __shared__ unsigned sIx[PCAP];
  __shared__ float    rS[256];
  __shared__ unsigned rI[256];
  __shared__ unsigned selP[P_];
  __shared__ unsigned selN[NNEG_];
  __shared__ int      sCnt[2];

  // ---- positives top-P ----
  int pcTrue = pC[b];
  int pc = min(pcTrue, PCAP);
  for (int i = t; i < pc; i += 256) { sS[i] = pS[b * PCAP + i]; sIx[i] = pI[b * PCAP + i]; }
  __syncthreads();
  int pos_num = topk_select(sS, sIx, pc, P_, rS, rI, selP, &sCnt[0]);  // == min(pcTrue, 64)
  __syncthreads();

  // ---- negatives top-NNEG ----
  int ncTot = nT[b];
  int ncl = min(nC[b], NCAP);
  for (int i = t; i < ncl; i += 256) { sS[i] = nS[b * NCAP + i]; sIx[i] = nI[b * NCAP + i]; }
  __syncthreads();
  int selNv = topk_select(sS, sIx, ncl, NNEG_, rS, rI, selN, &sCnt[1]);
  int neg_num = min(min(NNEG_, T_ - pos_num), ncTot);
  neg_num = min(neg_num, selNv);
  __syncthreads();

  // ---- per-sample rows ----
  if (t < T_) {
    float d0 = 0, d1 = 0, d2 = 0, c0 = 0, c1 = 0;
    int i0 = 0, i1 = 0;
    bool isp = t < pos_num;
    bool isn = !isp && (t < pos_num + neg_num);
    if (isp) {
      unsigned pk = selP[t];
      int g = (int)(pk / (unsigned)A_);
      int a = (int)(pk % (unsigned)A_);
      const float* pa = anch + ((size_t)b * A_ + a) * 4;
      float ay1 = pa[0], ay2 = pa[2];
      const float* gr = gtB + ((size_t)b * G_ + g) * 5;
      float gy1 = gr[0], gy2 = gr[2];
      float h = ay2 - ay1;
      float dy = (gy2 + gy1 - (ay2 + ay1)) * 0.5f / h;
      float dh = logf((gy2 - gy1) / h);
      d0 = dy * 10.f; d1 = dh * 5.f; d2 = 1.f;
      c0 = gtC[((size_t)b * G_ + g) * 2 + 0]; c1 = 1.f;
      i0 = vIdx[(size_t)b * A_ + a]; i1 = 1;
    } else if (isn) {
      i0 = (int)selN[t - pos_num]; i1 = -1;
      d2 = 1.f; c1 = 1.f;
    }
    float* oD = out + OFF_D + ((size_t)b * T_ + t) * 3;
    oD[0] = d0; oD[1] = d1; oD[2] = d2;
    float* oC = out + OFF_C + ((size_t)b * T_ + t) * 2;
    oC[0] = c0; oC[1] = c1;
    int* oI = (int*)(out + OFF_I) + ((size_t)b * T_ + t) * 2;
    oI[0] = i0; oI[1] = i1;
  }

  // ---- side deltas / indices per GT ----
  if (t < G_) {
    const float* gr = gtB + ((size_t)b * G_ + t) * 5;
    float tg = (gr[4] > 0.f) ? 1.f : 0.f;
    int tgi = (tg > 0.f) ? 1 : 0;
    unsigned li = (unsigned)(lE[b * 64 + t] & 0xFFFFFFFFull); if (li >= (unsigned)A_) li = 0;
    unsigned ri = 0xFFFFFFFFu - (unsigned)(rE[b * 64 + t] & 0xFFFFFFFFull); if (ri >= (unsigned)A_) ri = 0;
    const float* la = anch + ((size_t)b * A_ + li) * 4;
    const float* ra = anch + ((size_t)b * A_ + ri) * 4;
    float ld = (gr[1] - (la[3] + la[1]) * 0.5f) / (la[3] - la[1]) * 10.f;
    float rd = (gr[3] - (ra[3] + ra[1]) * 0.5f) / (ra[3] - ra[1]) * 10.f;
    float* oSD = out + OFF_SD + ((size_t)b * G_ + t) * 3;
    oSD[0] = ld * tg; oSD[1] = rd * tg; oSD[2] = tg;
    int* oSI = (int*)(out + OFF_SI) + ((size_t)b * G_ + t) * 3;
    oSI[0] = vIdx[(size_t)b * A_ + li] * tgi;
    oSI[1] = vIdx[(size_t)b * A_ + ri] * tgi;
    oSI[2] = tgi;
  }

  if (t == 0) {
    float gn = 0.f;
    for (int g = 0; g < G_; ++g) gn += (gtB[((size_t)b * G_ + g) * 5 + 4] > 0.f) ? 1.f : 0.f;
    out[OFF_GN + b] = gn;
    out[OFF_PN + b] = (float)pos_num;
    out[OFF_NN + b] = (float)neg_num;
  }
}

// =====================================================================
extern "C" void kernel_launch(void* const* d_in, const int* in_sizes, int n_in,
                              void* d_out, int out_size, void* d_ws, size_t ws_size,
                              hipStream_t stream) {
  const float* gtB  = (const float*)d_in[0];  // (B,G,5)
  const float* gtC  = (const float*)d_in[1];  // (B,G,2)
  const float* anch = (const float*)d_in[2];  // (B,A,4)
  const int*   vIdx = (const int*)d_in[3];    // (B,A)
  float* out = (float*)d_out;
  char* ws = (char*)d_ws;

  unsigned long long* lE = (unsigned long long*)(ws + 0);        // B*64 u64
  unsigned long long* rE = (unsigned long long*)(ws + 8192);     // B*64 u64
  unsigned* gmax = (unsigned*)(ws + 16384);                      // B*64 u32
  int* pC = (int*)(ws + 20480);
  int* nC = (int*)(ws + 20544);
  int* nT = (int*)(ws + 20608);
  float*    pS = (float*)(ws + 20672);
  unsigned* pI = (unsigned*)(ws + 20672 + 262144);
  float*    nS = (float*)(ws + 20672 + 524288);
  unsigned* nI = (unsigned*)(ws + 20672 + 786432);

  k_init<<<dim3(120), 256, 0, stream>>>(out, out_size, gmax, lE, rE, pC, nC, nT);
  k_pass1<<<dim3(B_, A_ / BLKA), 256, 0, stream>>>(gtB, anch, gmax);
  k_pass2<<<dim3(B_, A_ / BLKA), 256, 0, stream>>>(gtB, anch, gmax, lE, rE,
                                                   pC, pS, pI, nC, nT, nS, nI);
  k_final<<<dim3(B_), 256, 0, stream>>>(gtB, gtC, anch, vIdx, lE, rE,
                                        pC, pS, pI, nC, nT, nS, nI, out);
}